// TripletLoss_81595788689573
// MI455X (gfx1250) — compile-verified
//
#include <hip/hip_runtime.h>
#include <math.h>

// Problem constants (from reference: B=8192, C=1024)
#define B_SIZE 8192
#define C_DIM  1024
#define MARGIN 0.2f
#define EPSF   1e-6f
#define K_OVERRIDE 81   // max(2, int(8192*0.01))

typedef __attribute__((ext_vector_type(2))) float v2f;
typedef __attribute__((ext_vector_type(8))) float v8f;

// ---------------------------------------------------------------------------
// Kernel A: all_same flag (single block, deterministic tree reduction)
// ---------------------------------------------------------------------------
__global__ __launch_bounds__(1024) void all_same_kernel(const int* __restrict__ label,
                                                        int* __restrict__ flag) {
  __shared__ int sh[1024];
  const int tid = threadIdx.x;
  const int l0 = label[0];
  int ok = 1;
  const int base = tid * (B_SIZE / 1024);
#pragma unroll
  for (int j = 0; j < B_SIZE / 1024; ++j) ok &= (label[base + j] == l0);
  sh[tid] = ok;
  __syncthreads();
  for (int s = 512; s > 0; s >>= 1) {
    if (tid < s) sh[tid] &= sh[tid + s];
    __syncthreads();
  }
  if (tid == 0) flag[0] = sh[0];
}

__device__ __forceinline__ int eff_label(const int* __restrict__ label, int j, int all_same) {
  const int l = label[j];
  return (all_same && (j < K_OVERRIDE)) ? -1 : l;
}

// ---------------------------------------------------------------------------
// Kernel B: triplet mining — first positive / first negative per anchor
// ---------------------------------------------------------------------------
__global__ __launch_bounds__(256) void mine_kernel(const int* __restrict__ label,
                                                   const int* __restrict__ idx,
                                                   const int* __restrict__ flag,
                                                   int* __restrict__ pos,
                                                   int* __restrict__ neg,
                                                   int* __restrict__ valid) {
  const int i = blockIdx.x * blockDim.x + threadIdx.x;
  if (i >= B_SIZE) return;
  const int all_same = flag[0];
  const int li = eff_label(label, i, all_same);
  const int myidx = idx[i];
  int p = -1, n = -1;
  for (int j = 0; j < B_SIZE; ++j) {
    const int lj = eff_label(label, j, all_same);
    if (p < 0 && lj == li && idx[j] != myidx) p = j;
    if (n < 0 && lj != li) n = j;
    if (p >= 0 && n >= 0) break;
  }
  valid[i] = (p >= 0) && (n >= 0);
  pos[i] = (p < 0) ? 0 : p;   // argmax of all-false mask -> 0 (masked by valid)
  neg[i] = (n < 0) ? 0 : n;
}

// ---------------------------------------------------------------------------
// Kernel C: squared distances via V_WMMA_F32_16X16X4_F32 (D = Delta * Delta^T,
// diagonal = sum of squares). One wave handles 16 anchors; 256 thr = 8 waves.
//
// A-matrix f32 16x4 layout: lane l, VGPR v -> A[l%16][2*(l/16)+v]. For
// B = A^T (4x16) the per-lane value at (lane l, VGPR v) is A[l%16][2*(l/16)+v]
// as well, so the same register pair serves as both A and B operands.
// ---------------------------------------------------------------------------
__global__ __launch_bounds__(256) void dist_wmma_kernel(const float* __restrict__ z,
                                                        const int* __restrict__ pos,
                                                        const int* __restrict__ neg,
                                                        const int* __restrict__ valid,
                                                        float* __restrict__ per_out) {
  const int tid  = threadIdx.x;
  const int wave = tid >> 5;
  const int lane = tid & 31;
  const int m    = lane & 15;   // anchor row within tile
  const int half = lane >> 4;   // k-half selector
  const int base = blockIdx.x * 128 + wave * 16;
  const int r    = base + m;

  const float* __restrict__ za = z + (size_t)r * C_DIM;
  const float* __restrict__ zp = z + (size_t)pos[r] * C_DIM;
  const float* __restrict__ zn = z + (size_t)neg[r] * C_DIM;
  const int koff = half * 2;

  v8f cp = {};
  v8f cn = {};
#pragma unroll 4
  for (int kk = 0; kk < C_DIM; kk += 4) {
    const int kb = kk + koff;
    const v2f a = *(const v2f*)(za + kb);
    const v2f p = *(const v2f*)(zp + kb);
    const v2f n = *(const v2f*)(zn + kb);
    const v2f dp = a - p + EPSF;   // reference: (a - b + EPS)
    const v2f dn = a - n + EPSF;
    // D += Delta * Delta^T  (diagonal accumulates squared L2 norms)
    cp = __builtin_amdgcn_wmma_f32_16x16x4_f32(false, dp, false, dp, (short)0, cp, false, false);
    cn = __builtin_amdgcn_wmma_f32_16x16x4_f32(false, dn, false, dn, (short)0, cn, false, false);
  }

  // Diagonal extraction per C/D layout:
  //   i in 0..7  -> lane i,      VGPR i
  //   i in 8..15 -> lane i + 16, VGPR i - 8
  const bool diag = (lane < 8) || (lane >= 24);
  const int vsel  = (lane < 8) ? lane : (lane - 24);
  const int ai    = (lane < 8) ? lane : (lane - 16);
  float dap2 = 0.f, dan2 = 0.f;
#pragma unroll
  for (int v = 0; v < 8; ++v) {
    dap2 = (v == vsel) ? cp[v] : dap2;
    dan2 = (v == vsel) ? cn[v] : dan2;
  }
  if (diag) {
    const int rr = base + ai;
    const float d_ap = sqrtf(dap2);
    const float d_an = sqrtf(dan2);
    const float per = fmaxf(d_ap - d_an + MARGIN, 0.f);
    per_out[rr] = per * (float)valid[rr];
  }
}

// ---------------------------------------------------------------------------
// Kernel D: deterministic final reduction -> scalar loss
// ---------------------------------------------------------------------------
__global__ __launch_bounds__(256) void finalize_kernel(const float* __restrict__ per,
                                                       const int* __restrict__ valid,
                                                       float* __restrict__ out) {
  __shared__ float sf[256];
  __shared__ int   si[256];
  const int tid = threadIdx.x;
  float s = 0.f;
  int nv = 0;
  for (int i = tid; i < B_SIZE; i += 256) {  // fixed traversal order
    s += per[i];
    nv += valid[i];
  }
  sf[tid] = s;
  si[tid] = nv;
  __syncthreads();
  for (int st = 128; st > 0; st >>= 1) {
    if (tid < st) { sf[tid] += sf[tid + st]; si[tid] += si[tid + st]; }
    __syncthreads();
  }
  if (tid == 0) {
    const int n = si[0];
    out[0] = (n > 0) ? (sf[0] / (float)((n < 1) ? 1 : n)) : 0.f;
  }
}

// ---------------------------------------------------------------------------
// Launch
// ---------------------------------------------------------------------------
extern "C" void kernel_launch(void* const* d_in, const int* in_sizes, int n_in,
                              void* d_out, int out_size, void* d_ws, size_t ws_size,
                              hipStream_t stream) {
  (void)in_sizes; (void)n_in; (void)out_size; (void)ws_size;

  const int*   z_label = (const int*)d_in[0];
  const int*   z_idx   = (const int*)d_in[1];
  const float* z       = (const float*)d_in[2];
  float*       out     = (float*)d_out;

  // Workspace layout: pos[B], neg[B], valid[B] (int), per[B] (float), flag (int)
  int*   ws_pos   = (int*)d_ws;
  int*   ws_neg   = ws_pos + B_SIZE;
  int*   ws_valid = ws_neg + B_SIZE;
  float* ws_per   = (float*)(ws_valid + B_SIZE);
  int*   ws_flag  = (int*)(ws_per + B_SIZE);

  all_same_kernel<<<1, 1024, 0, stream>>>(z_label, ws_flag);
  mine_kernel<<<B_SIZE / 256, 256, 0, stream>>>(z_label, z_idx, ws_flag,
                                                ws_pos, ws_neg, ws_valid);
  dist_wmma_kernel<<<B_SIZE / 128, 256, 0, stream>>>(z, ws_pos, ws_neg, ws_valid, ws_per);
  finalize_kernel<<<1, 256, 0, stream>>>(ws_per, ws_valid, out);
}